// SNEA_87625922773403
// MI455X (gfx1250) — compile-verified
//
#include <hip/hip_runtime.h>
#include <math.h>

typedef float v2f __attribute__((ext_vector_type(2)));
typedef float v8f __attribute__((ext_vector_type(8)));

// ---------------------------------------------------------------------------
// Utility kernels
// ---------------------------------------------------------------------------
__global__ void fill_f32(float* __restrict__ p, float v, long long n) {
    long long i = (long long)blockIdx.x * blockDim.x + threadIdx.x;
    long long stride = (long long)gridDim.x * blockDim.x;
    for (; i < n; i += stride) p[i] = v;
}

__device__ __forceinline__ void atomic_max_f32(float* addr, float val) {
    unsigned int* ua = (unsigned int*)addr;
    unsigned int old = __float_as_uint(*addr);
    while (__uint_as_float(old) < val) {
        unsigned int prev = atomicCAS(ua, old, __float_as_uint(val));
        if (prev == old) break;
        old = prev;
    }
}

// ---------------------------------------------------------------------------
// Attention aggregation passes
// ---------------------------------------------------------------------------
// si[i] = z[i,:] . alpha[:d]   sj[i] = z[i,:] . alpha[d:]
__global__ void node_scores(const float* __restrict__ z, int d,
                            const float* __restrict__ alpha,
                            float* __restrict__ si, float* __restrict__ sj, int n) {
    int i = blockIdx.x * blockDim.x + threadIdx.x;
    if (i >= n) return;
    const float* zr = z + (size_t)i * d;
    float a = 0.f, b = 0.f;
    for (int k = 0; k < d; ++k) {
        float v = zr[k];
        a += v * alpha[k];
        b += v * alpha[d + k];
    }
    si[i] = a;
    sj[i] = b;
}

// score[e] = si[dst] + sj[src];  m[dst] = max(m[dst], score)
__global__ void edge_score_max(const int* __restrict__ src, const int* __restrict__ dst,
                               const float* __restrict__ si, const float* __restrict__ sj,
                               float* __restrict__ score, float* __restrict__ m, int E) {
    int e = blockIdx.x * blockDim.x + threadIdx.x;
    if (e >= E) return;
    int dn = dst[e];
    float sc = si[dn] + sj[src[e]];
    score[e] = sc;
    atomic_max_f32(&m[dn], sc);
}

// ex[e] = exp(score[e] - m[dst]);  denom[dst] += ex[e]   (ex overwrites score in place)
__global__ void edge_exp_sum(const int* __restrict__ dst,
                             float* __restrict__ score_ex,
                             const float* __restrict__ m,
                             float* __restrict__ denom, int E) {
    int e = blockIdx.x * blockDim.x + threadIdx.x;
    if (e >= E) return;
    int dn = dst[e];
    float ex = expf(score_ex[e] - m[dn]);
    score_ex[e] = ex;
    atomicAdd(&denom[dn], ex);
}

// agg[dst,k] += (ex[e]/max(denom[dst],1e-16)) * z[src,k]
// One thread per (edge,feature); d = 1<<ld, so a wave32 shares one edge's
// broadcast scalars and does coalesced gathers / atomic adds on features.
__global__ void edge_scatter(const int* __restrict__ src, const int* __restrict__ dst,
                             const float* __restrict__ ex, const float* __restrict__ denom,
                             const float* __restrict__ z, int ld,
                             float* __restrict__ agg, int E) {
    long long t = (long long)blockIdx.x * blockDim.x + threadIdx.x;
    int e = (int)(t >> ld);
    if (e >= E) return;
    int k = (int)(t & ((1 << ld) - 1));
    int dn = dst[e];
    float w = ex[e] / fmaxf(denom[dn], 1e-16f);
    atomicAdd(&agg[((size_t)dn << ld) + k], w * z[((size_t)src[e] << ld) + k]);
}

// ---------------------------------------------------------------------------
// Fused WMMA GEMM:  out[:, off:off+32] = tanh( [A0|A1|A2] @ W + bias )
// W is (D0+D1+D2) x 32 row-major. One wave32 -> one 16x16 output tile using
// V_WMMA_F32_16X16X4_F32 (exact fp32 math, matches reference numerics).
// Part widths are template constants so the K loop fully unrolls into a
// straight-line WMMA chain with b64 A-fragment loads, no per-trip branching.
// Fragment layouts per CDNA5 ISA 7.12.2 (32-bit A 16x4 / B 4x16 / C 16x16).
// ---------------------------------------------------------------------------
template <int D0, int D1, int D2>
__global__ __launch_bounds__(256)
void wmma_gemm_tanh_t(const float* __restrict__ A0,
                      const float* __restrict__ A1,
                      const float* __restrict__ A2,
                      const float* __restrict__ W,
                      const float* __restrict__ bias,
                      float* __restrict__ out, int out_stride, int out_col_off,
                      int n) {
    const int lane = threadIdx.x & 31;
    const int wave = threadIdx.x >> 5;
    const int row0 = (blockIdx.x * 4 + (wave >> 1)) * 16;  // row tile (16 rows)
    if (row0 >= n) return;                // uniform per wave: EXEC stays all-1s
    const int colbase = (wave & 1) * 16;  // 2 col tiles cover the 32 output cols
    const int mrow = lane & 15;
    const int khalf = (lane >> 4) << 1;   // lanes 16-31 hold K+2,K+3
    const int col = colbase + mrow;
    int row = row0 + mrow;
    if (row >= n) row = n - 1;            // safety clamp (n % 16 == 0 in practice)

    v8f c = {0.f, 0.f, 0.f, 0.f, 0.f, 0.f, 0.f, 0.f};

#pragma unroll
    for (int k = 0; k < D0; k += 4) {
        v2f a = *(const v2f*)(A0 + (size_t)row * D0 + k + khalf);
        v2f b;
        b.x = W[(size_t)(k + khalf) * 32 + col];
        b.y = W[(size_t)(k + khalf + 1) * 32 + col];
        c = __builtin_amdgcn_wmma_f32_16x16x4_f32(false, a, false, b,
                                                  (short)0, c, false, false);
    }
#pragma unroll
    for (int k = 0; k < D1; k += 4) {
        v2f a = *(const v2f*)(A1 + (size_t)row * D1 + k + khalf);
        v2f b;
        b.x = W[(size_t)(D0 + k + khalf) * 32 + col];
        b.y = W[(size_t)(D0 + k + khalf + 1) * 32 + col];
        c = __builtin_amdgcn_wmma_f32_16x16x4_f32(false, a, false, b,
                                                  (short)0, c, false, false);
    }
#pragma unroll
    for (int k = 0; k < D2; k += 4) {
        v2f a = *(const v2f*)(A2 + (size_t)row * D2 + k + khalf);
        v2f b;
        b.x = W[(size_t)(D0 + D1 + k + khalf) * 32 + col];
        b.y = W[(size_t)(D0 + D1 + k + khalf + 1) * 32 + col];
        c = __builtin_amdgcn_wmma_f32_16x16x4_f32(false, a, false, b,
                                                  (short)0, c, false, false);
    }

    const float bv = bias[col];
    const int mbase = row0 + ((lane >> 4) << 3);  // C/D: lanes 16-31 hold M=v+8
#pragma unroll
    for (int v = 0; v < 8; ++v) {
        int r = mbase + v;
        if (r < n) out[(size_t)r * out_stride + out_col_off + col] = tanhf(c[v] + bv);
    }
}

// ---------------------------------------------------------------------------
// Launcher
// ---------------------------------------------------------------------------
static inline int cdiv(long long a, int b) { return (int)((a + b - 1) / b); }

extern "C" void kernel_launch(void* const* d_in, const int* in_sizes, int n_in,
                              void* d_out, int out_size, void* d_ws, size_t ws_size,
                              hipStream_t stream) {
    const float* x        = (const float*)d_in[0];
    const int*   pei      = (const int*)d_in[1];
    const int*   nei      = (const int*)d_in[2];
    const float* alpha1_b = (const float*)d_in[3];
    const float* alpha1_u = (const float*)d_in[4];
    const float* W1_b     = (const float*)d_in[5];
    const float* b1_b     = (const float*)d_in[6];
    const float* W1_u     = (const float*)d_in[7];
    const float* b1_u     = (const float*)d_in[8];
    const float* alpha2_b = (const float*)d_in[9];
    const float* alpha2_u = (const float*)d_in[10];
    const float* W2_b     = (const float*)d_in[11];
    const float* b2_b     = (const float*)d_in[12];
    const float* W2_u     = (const float*)d_in[13];
    const float* b2_u     = (const float*)d_in[14];

    const int N  = in_sizes[0] / 64;
    const int EP = in_sizes[1] / 2;
    const int EN = in_sizes[2] / 2;
    const int* ps = pei;  const int* pd = pei + EP;
    const int* ns = nei;  const int* nd = nei + EN;

    // Workspace layout (floats)
    float* f    = (float*)d_ws;
    float* zb   = f;  f += (size_t)N * 32;
    float* zu   = f;  f += (size_t)N * 32;
    float* aggA = f;  f += (size_t)N * 64;
    float* aggB = f;  f += (size_t)N * 32;
    float* si   = f;  f += N;
    float* sj   = f;  f += N;
    float* mbuf = f;  f += N;
    float* dnm  = f;  f += N;
    float* ebuf = f;  // E-sized (max EP)
    float* outp = (float*)d_out;

    auto run_agg = [&](const float* z, int d, int ld, const float* alpha,
                       const int* src, const int* dst, int E, float* agg) {
        fill_f32<<<256, 256, 0, stream>>>(mbuf, -INFINITY, N);
        fill_f32<<<256, 256, 0, stream>>>(dnm, 0.f, N);
        fill_f32<<<2048, 256, 0, stream>>>(agg, 0.f, (long long)N * d);
        node_scores<<<cdiv(N, 256), 256, 0, stream>>>(z, d, alpha, si, sj, N);
        edge_score_max<<<cdiv(E, 256), 256, 0, stream>>>(src, dst, si, sj, ebuf, mbuf, E);
        edge_exp_sum<<<cdiv(E, 256), 256, 0, stream>>>(dst, ebuf, mbuf, dnm, E);
        edge_scatter<<<cdiv((long long)E * d, 256), 256, 0, stream>>>(src, dst, ebuf, dnm, z, ld, agg, E);
    };

    const int gblocks = cdiv(N, 64);  // 4 row-tiles (x 2 col-tiles) per 256-thread block

    // ---- Layer 1 ----
    run_agg(x, 64, 6, alpha1_b, ps, pd, EP, aggA);
    wmma_gemm_tanh_t<64, 64, 0><<<gblocks, 256, 0, stream>>>(
        aggA, x, nullptr, W1_b, b1_b, zb, 32, 0, N);
    run_agg(x, 64, 6, alpha1_u, ns, nd, EN, aggA);
    wmma_gemm_tanh_t<64, 64, 0><<<gblocks, 256, 0, stream>>>(
        aggA, x, nullptr, W1_u, b1_u, zu, 32, 0, N);

    // ---- Layer 2 ----
    run_agg(zb, 32, 5, alpha2_b, ps, pd, EP, aggA);   // agg_bp
    run_agg(zu, 32, 5, alpha2_b, ns, nd, EN, aggB);   // agg_bn
    wmma_gemm_tanh_t<32, 32, 32><<<gblocks, 256, 0, stream>>>(
        aggA, aggB, zb, W2_b, b2_b, outp, 64, 0, N);
    run_agg(zu, 32, 5, alpha2_u, ps, pd, EP, aggA);   // agg_up
    run_agg(zb, 32, 5, alpha2_u, ns, nd, EN, aggB);   // agg_un
    wmma_gemm_tanh_t<32, 32, 32><<<gblocks, 256, 0, stream>>>(
        aggA, aggB, zu, W2_u, b2_u, outp, 64, 32, N);
}